// PointNetSetAbstraction_64596308132330
// MI455X (gfx1250) — compile-verified
//
#include <hip/hip_runtime.h>
#include <hip/hip_bf16.h>
#include <math.h>

typedef __attribute__((ext_vector_type(2))) float v2f;
typedef __attribute__((ext_vector_type(8))) float v8f;

__device__ __forceinline__ float gelu_f(float x) {
    return 0.5f * x * (1.0f + erff(x * 0.70710678118654752f));
}

// ---------------------------------------------------------------------------
// 1) Farthest point sampling: grid=B, block=256. Writes new_xyz into d_out.
// ---------------------------------------------------------------------------
__global__ __launch_bounds__(256) void fps_kernel(const float* __restrict__ xyz,
                                                  float* __restrict__ newxyz) {
    const int b = blockIdx.x, t = threadIdx.x;
    const size_t base = (size_t)b * 4096 * 3;
    float px[16], py[16], pz[16], dist[16];
#pragma unroll
    for (int i = 0; i < 16; ++i) {
        int j = t + i * 256;
        px[i] = xyz[base + (size_t)j * 3 + 0];
        py[i] = xyz[base + (size_t)j * 3 + 1];
        pz[i] = xyz[base + (size_t)j * 3 + 2];
        dist[i] = 1e10f;
    }
    __shared__ float s_red[256];
    __shared__ int s_idx[256];
    __shared__ float s_c[3];
    __shared__ int s_far;
    if (t == 0) s_far = 0;
    __syncthreads();
    for (int it = 0; it < 1024; ++it) {
        if (t == 0) {
            int far = s_far;
            float cx = xyz[base + (size_t)far * 3 + 0];
            float cy = xyz[base + (size_t)far * 3 + 1];
            float cz = xyz[base + (size_t)far * 3 + 2];
            s_c[0] = cx; s_c[1] = cy; s_c[2] = cz;
            size_t o = ((size_t)b * 1024 + it) * 3;
            newxyz[o + 0] = cx; newxyz[o + 1] = cy; newxyz[o + 2] = cz;
        }
        __syncthreads();
        float cx = s_c[0], cy = s_c[1], cz = s_c[2];
        float bv = -1.0f; int bi = 0;
#pragma unroll
        for (int i = 0; i < 16; ++i) {
            float dx = px[i] - cx, dy = py[i] - cy, dz = pz[i] - cz;
            float d = dx * dx + dy * dy + dz * dz;
            dist[i] = fminf(dist[i], d);
            if (dist[i] > bv) { bv = dist[i]; bi = t + i * 256; }
        }
        s_red[t] = bv; s_idx[t] = bi;
        __syncthreads();
        for (int off = 128; off > 0; off >>= 1) {
            if (t < off) {
                float ov = s_red[t + off]; int oi = s_idx[t + off];
                if (ov > s_red[t] || (ov == s_red[t] && oi < s_idx[t])) {
                    s_red[t] = ov; s_idx[t] = oi;
                }
            }
            __syncthreads();
        }
        if (t == 0) s_far = s_idx[0];
        __syncthreads();
    }
}

// ---------------------------------------------------------------------------
// 2) t-embedding projection: tvec[layer][b][oc] = gelu(t_embed[b]) @ tw.T + tb
//    grid=B (16), block=256 (64+64+128 outputs per batch).
// ---------------------------------------------------------------------------
__global__ __launch_bounds__(256) void tproj_kernel(
    const float* __restrict__ te,
    const float* __restrict__ tw0, const float* __restrict__ tb0,
    const float* __restrict__ tw1, const float* __restrict__ tb1,
    const float* __restrict__ tw2, const float* __restrict__ tb2,
    float* __restrict__ tvec) {
    const int b = blockIdx.x, t = threadIdx.x;
    __shared__ float g[128];
    if (t < 128) g[t] = gelu_f(te[b * 128 + t]);
    __syncthreads();
    int oc, Co, tbase;
    const float* tw; const float* tb;
    if (t < 64)       { oc = t;       Co = 64;  tw = tw0; tb = tb0; tbase = 0;    }
    else if (t < 128) { oc = t - 64;  Co = 64;  tw = tw1; tb = tb1; tbase = 1024; }
    else              { oc = t - 128; Co = 128; tw = tw2; tb = tb2; tbase = 2048; }
    float acc = tb[oc];
    for (int j = 0; j < 128; ++j) acc += g[j] * tw[oc * 128 + j];
    tvec[tbase + b * Co + oc] = acc;
}

// ---------------------------------------------------------------------------
// 3) Ball query: per-thread 32-entry max-heap in LDS, xyz staged in chunks.
//    grid=(8,16), block=128. Heap layout hd[k*128+t] (bank-conflict free).
// ---------------------------------------------------------------------------
__device__ __forceinline__ void heap_sift(float* hd, int* hi, int t, int i) {
    for (;;) {
        int l = 2 * i + 1;
        if (l >= 32) break;
        int r = l + 1, c = l;
        if (r < 32 && hd[r * 128 + t] > hd[l * 128 + t]) c = r;
        if (hd[c * 128 + t] > hd[i * 128 + t]) {
            float td = hd[c * 128 + t]; hd[c * 128 + t] = hd[i * 128 + t]; hd[i * 128 + t] = td;
            int ti = hi[c * 128 + t]; hi[c * 128 + t] = hi[i * 128 + t]; hi[i * 128 + t] = ti;
            i = c;
        } else break;
    }
}

__global__ __launch_bounds__(128) void ballq_kernel(const float* __restrict__ xyz,
                                                    const float* __restrict__ newxyz,
                                                    int* __restrict__ gidx) {
    const int t = threadIdx.x;
    const int b = blockIdx.y;
    const int s = blockIdx.x * 128 + t;
    __shared__ float sx[2048], sy[2048], sz[2048];
    __shared__ float hd[32 * 128];
    __shared__ int hi[32 * 128];
    const size_t qo = ((size_t)b * 1024 + s) * 3;
    const float qx = newxyz[qo], qy = newxyz[qo + 1], qz = newxyz[qo + 2];
    const size_t base = (size_t)b * 4096 * 3;
    int cnt = 0;
    for (int ch = 0; ch < 2; ++ch) {
        __syncthreads();
        for (int i = t; i < 2048; i += 128) {
            size_t o = base + (size_t)(ch * 2048 + i) * 3;
            sx[i] = xyz[o]; sy[i] = xyz[o + 1]; sz[i] = xyz[o + 2];
        }
        __syncthreads();
        for (int j = 0; j < 2048; ++j) {
            float dx = sx[j] - qx, dy = sy[j] - qy, dz = sz[j] - qz;
            float d = dx * dx + dy * dy + dz * dz;
            int jj = ch * 2048 + j;
            if (cnt < 32) {
                hd[cnt * 128 + t] = d; hi[cnt * 128 + t] = jj;
                ++cnt;
                if (cnt == 32)
                    for (int i2 = 15; i2 >= 0; --i2) heap_sift(hd, hi, t, i2);
            } else if (d < hd[t]) {
                hd[t] = d; hi[t] = jj;
                heap_sift(hd, hi, t, 0);
            }
        }
    }
    float bd = hd[t]; int bi = hi[t];
    for (int k = 1; k < 32; ++k) {
        float d = hd[k * 128 + t]; int ix = hi[k * 128 + t];
        if (d < bd || (d == bd && ix < bi)) { bd = d; bi = ix; }
    }
    int* gp = gidx + (((size_t)b * 1024 + s) << 5);
    for (int k = 0; k < 32; ++k) {
        float d = hd[k * 128 + t]; int ix = hi[k * 128 + t];
        gp[k] = (d <= 0.04f) ? ix : bi;  // radius^2 = 0.04
    }
}

// ---------------------------------------------------------------------------
// 4) Fused GEMM (+ gather / + normalize-GELU of prev layer) + bias + BN stats
//    D = W (COUTxCIN) * X (CINx524288). WMMA f32 16x16x4.
//    block=128 threads (4 waves), N-tile=64, M-tile=COUT, full K per block.
// ---------------------------------------------------------------------------
template <int MODE, int CIN, int COUT>
__global__ __launch_bounds__(128) void gemm_bn_kernel(
    const float* __restrict__ W, const float* __restrict__ bias,
    const float* __restrict__ xin, const float* __restrict__ scale_p,
    const float* __restrict__ shift_p,
    const float* __restrict__ xyz, const float* __restrict__ points,
    const int* __restrict__ gidx, const float* __restrict__ newxyz,
    float* __restrict__ yout, float* __restrict__ gsum, float* __restrict__ gsqs) {
    constexpr int CPAD = (MODE == 0) ? 68 : CIN;  // K extent (padded, mult of 4)
    constexpr int KLD = 68;                        // sW leading dim (bank-safe)
    constexpr int NT = 64;
    constexpr int MT = COUT / 16;
    constexpr int KSTEPS = CPAD / 4;

    const int t = threadIdx.x;
    const int lane = t & 31, wave = t >> 5;
    const int b = blockIdx.y;
    const int Lbase = blockIdx.x * NT;

    __shared__ float sW[COUT * KLD];
    __shared__ float sX[68 * NT];
    __shared__ float sBias[COUT];
    __shared__ float sSum[COUT];
    __shared__ float sSqs[COUT];
    __shared__ int sgid[NT];
    __shared__ float sq[3][NT];

    for (int e = t; e < COUT * KLD; e += 128) {
        int m = e / KLD, k = e - m * KLD;
        sW[e] = (k < CIN) ? W[m * CIN + k] : 0.0f;
    }
    if (t < COUT) { sBias[t] = bias[t]; sSum[t] = 0.0f; sSqs[t] = 0.0f; }
    if constexpr (MODE == 0) {
        if (t < NT) {
            int l = Lbase + t;
            int s = l >> 5, kk = l & 31;
            sgid[t] = gidx[(((size_t)b << 10) + s) * 32 + kk];
            size_t qo = (((size_t)b << 10) + s) * 3;
            sq[0][t] = newxyz[qo]; sq[1][t] = newxyz[qo + 1]; sq[2][t] = newxyz[qo + 2];
        }
    }
    __syncthreads();

    if constexpr (MODE == 0) {
        for (int e = t; e < 68 * NT; e += 128) {
            int c = e >> 6, n = e & 63;
            float v;
            if (c < 3)
                v = xyz[((size_t)b * 4096 + sgid[n]) * 3 + c] - sq[c][n];
            else if (c < 67)
                v = points[((size_t)b * 64 + (c - 3)) * 4096 + sgid[n]];
            else
                v = 0.0f;
            sX[e] = v;
        }
    } else {
        for (int e = t; e < CIN * NT; e += 128) {
            int c = e >> 6, n = e & 63;
            float yv = xin[((size_t)(b * CIN + c)) * 32768 + Lbase + n];
            sX[e] = gelu_f(yv * scale_p[c] + shift_p[b * 128 + c]);
        }
    }
    __syncthreads();

    const int n0 = wave * 16;
    const int lm = lane & 15, lh = lane >> 4;
    v8f acc[MT];
    const v8f zacc = {0.f, 0.f, 0.f, 0.f, 0.f, 0.f, 0.f, 0.f};
#pragma unroll
    for (int mt = 0; mt < MT; ++mt) acc[mt] = zacc;

#pragma unroll
    for (int ks = 0; ks < KSTEPS; ++ks) {
        const int k0 = ks * 4 + lh * 2;  // A/B K split across half-waves
        v2f bf;
        bf.x = sX[k0 * NT + n0 + lm];
        bf.y = sX[(k0 + 1) * NT + n0 + lm];
#pragma unroll
        for (int mt = 0; mt < MT; ++mt) {
            const int m = mt * 16 + lm;
            v2f af;
            af.x = sW[m * KLD + k0];
            af.y = sW[m * KLD + k0 + 1];
            acc[mt] = __builtin_amdgcn_wmma_f32_16x16x4_f32(
                false, af, false, bf, (short)0, acc[mt], false, false);
        }
    }

    // epilogue: bias, store y, accumulate per-channel sum / sum-of-squares
    const int col = Lbase + n0 + lm;
#pragma unroll
    for (int mt = 0; mt < MT; ++mt) {
#pragma unroll
        for (int v = 0; v < 8; ++v) {
            const int m = mt * 16 + lh * 8 + v;  // D row = vgpr + 8*half
            float yv = acc[mt][v] + sBias[m];
            yout[((size_t)(b * COUT + m)) * 32768 + col] = yv;
            float sv = yv, s2 = yv * yv;
#pragma unroll
            for (int off = 1; off < 16; off <<= 1) {  // reduce within half-wave
                sv += __shfl_xor(sv, off, 32);
                s2 += __shfl_xor(s2, off, 32);
            }
            if (lm == 0) {
                atomicAdd(&sSum[m], sv);
                atomicAdd(&sSqs[m], s2);
            }
        }
    }
    __syncthreads();
    for (int c = t; c < COUT; c += 128) {
        atomicAdd(&gsum[c], sSum[c]);
        atomicAdd(&gsqs[c], sSqs[c]);
    }
}

// ---------------------------------------------------------------------------
// 5) Finalize BN: fold (mu, var, gamma, beta, tvec) -> scale[c], shift[b][c]
// ---------------------------------------------------------------------------
__global__ void finalize_kernel(const float* __restrict__ gsum,
                                const float* __restrict__ gsqs,
                                const float* __restrict__ gamma,
                                const float* __restrict__ beta,
                                const float* __restrict__ tvec_l,
                                float* __restrict__ scale_l,
                                float* __restrict__ shift_l, int Cout) {
    int c = threadIdx.x;
    if (c >= Cout) return;
    const float inv = 1.0f / 524288.0f;  // B * L = 16 * 32768
    float mu = gsum[c] * inv;
    float var = gsqs[c] * inv - mu * mu;
    if (var < 0.0f) var = 0.0f;
    float istd = rsqrtf(var + 1e-5f);
    float sc = gamma[c] * istd;
    scale_l[c] = sc;
    float b0 = beta[c] - mu * sc;
    for (int b = 0; b < 16; ++b) shift_l[b * 128 + c] = b0 + tvec_l[b * Cout + c];
}

// ---------------------------------------------------------------------------
// 6) norm + GELU + max over K=32 -> new_points (B,128,S)
// ---------------------------------------------------------------------------
__global__ __launch_bounds__(256) void maxpool_kernel(const float* __restrict__ y3,
                                                      const float* __restrict__ scale_l,
                                                      const float* __restrict__ shift_l,
                                                      float* __restrict__ out) {
    int idx = blockIdx.x * 256 + threadIdx.x;  // 16*128*1024 total
    int b = idx >> 17;
    int rem = idx & 131071;
    int o = rem >> 10;
    int s = rem & 1023;
    const float4* p = (const float4*)(y3 + ((size_t)(b * 128 + o)) * 32768 + (size_t)s * 32);
    float sc = scale_l[o];
    float sh = shift_l[b * 128 + o];
    float m = -1e30f;
#pragma unroll
    for (int q = 0; q < 8; ++q) {
        float4 v = p[q];
        m = fmaxf(m, gelu_f(v.x * sc + sh));
        m = fmaxf(m, gelu_f(v.y * sc + sh));
        m = fmaxf(m, gelu_f(v.z * sc + sh));
        m = fmaxf(m, gelu_f(v.w * sc + sh));
    }
    out[49152 + idx] = m;
}

// ---------------------------------------------------------------------------
extern "C" void kernel_launch(void* const* d_in, const int* in_sizes, int n_in,
                              void* d_out, int out_size, void* d_ws, size_t ws_size,
                              hipStream_t stream) {
    (void)in_sizes; (void)n_in; (void)out_size; (void)ws_size;
    const float* xyz    = (const float*)d_in[0];
    const float* points = (const float*)d_in[1];
    const float* te     = (const float*)d_in[2];
    const float* W0 = (const float*)d_in[3];  const float* B0 = (const float*)d_in[4];
    const float* G0 = (const float*)d_in[5];  const float* Be0 = (const float*)d_in[6];
    const float* TW0 = (const float*)d_in[7]; const float* TB0 = (const float*)d_in[8];
    const float* W1 = (const float*)d_in[9];  const float* B1 = (const float*)d_in[10];
    const float* G1 = (const float*)d_in[11]; const float* Be1 = (const float*)d_in[12];
    const float* TW1 = (const float*)d_in[13]; const float* TB1 = (const float*)d_in[14];
    const float* W2 = (const float*)d_in[15]; const float* B2 = (const float*)d_in[16];
    const float* G2 = (const float*)d_in[17]; const float* Be2 = (const float*)d_in[18];
    const float* TW2 = (const float*)d_in[19]; const float* TB2 = (const float*)d_in[20];

    float* out = (float*)d_out;
    char* ws = (char*)d_ws;
    size_t off = 0;
    int* gidx = (int*)(ws + off);       off += (size_t)524288 * 4;
    float* tvec = (float*)(ws + off);   off += (size_t)4096 * 4;
    float* stats = (float*)(ws + off);  const size_t statsBytes = 3 * 2 * 128 * 4; off += statsBytes;
    float* scales = (float*)(ws + off); off += (size_t)3 * 128 * 4;
    float* shifts = (float*)(ws + off); off += (size_t)3 * 2048 * 4;
    off = (off + 255) & ~(size_t)255;
    float* y1 = (float*)(ws + off); off += (size_t)16 * 64 * 32768 * 4;
    float* y2 = (float*)(ws + off); off += (size_t)16 * 64 * 32768 * 4;
    float* y3 = (float*)(ws + off); off += (size_t)16 * 128 * 32768 * 4;
    float* newxyz = out;  // first 49152 floats of d_out

    hipMemsetAsync(stats, 0, statsBytes, stream);

    fps_kernel<<<16, 256, 0, stream>>>(xyz, newxyz);
    tproj_kernel<<<16, 256, 0, stream>>>(te, TW0, TB0, TW1, TB1, TW2, TB2, tvec);
    ballq_kernel<<<dim3(8, 16), 128, 0, stream>>>(xyz, newxyz, gidx);

    gemm_bn_kernel<0, 67, 64><<<dim3(512, 16), 128, 0, stream>>>(
        W0, B0, nullptr, nullptr, nullptr, xyz, points, gidx, newxyz,
        y1, stats + 0, stats + 128);
    finalize_kernel<<<1, 128, 0, stream>>>(stats + 0, stats + 128, G0, Be0,
                                           tvec + 0, scales + 0, shifts + 0, 64);

    gemm_bn_kernel<1, 64, 64><<<dim3(512, 16), 128, 0, stream>>>(
        W1, B1, y1, scales + 0, shifts + 0, nullptr, nullptr, nullptr, nullptr,
        y2, stats + 256, stats + 384);
    finalize_kernel<<<1, 128, 0, stream>>>(stats + 256, stats + 384, G1, Be1,
                                           tvec + 1024, scales + 128, shifts + 2048, 64);

    gemm_bn_kernel<2, 64, 128><<<dim3(512, 16), 128, 0, stream>>>(
        W2, B2, y2, scales + 128, shifts + 2048, nullptr, nullptr, nullptr, nullptr,
        y3, stats + 512, stats + 640);
    finalize_kernel<<<1, 128, 0, stream>>>(stats + 512, stats + 640, G2, Be2,
                                           tvec + 2048, scales + 256, shifts + 4096, 128);

    maxpool_kernel<<<8192, 256, 0, stream>>>(y3, scales + 256, shifts + 4096, out);
}